// TransitionDown_74440373174612
// MI455X (gfx1250) — compile-verified
//
#include <hip/hip_runtime.h>

typedef __attribute__((ext_vector_type(2))) float v2f;
typedef __attribute__((ext_vector_type(8))) float v8f;

#define C_IN   64
#define KNN    16
#define KDIM   68      // 64 feat + 3 xyz + 1 zero pad (K padded to 4*17)
#define KSTEPS (KDIM / 4)
#define C_OUT  128
#define SPW    32      // samples per workgroup (amortize W-frag load + stats atomics)
#define BN_EPS 1e-5f

// ---------------------------------------------------------------------------
// Phase 0: new_point = point[sample_idx]
// ---------------------------------------------------------------------------
__global__ __launch_bounds__(256) void td_newpoint(const float* __restrict__ point,
                                                   const int* __restrict__ sample_idx,
                                                   float* __restrict__ out, int m) {
    int i = blockIdx.x * blockDim.x + threadIdx.x;
    if (i < m) {
        int s = sample_idx[i];
        out[i * 3 + 0] = point[s * 3 + 0];
        out[i * 3 + 1] = point[s * 3 + 1];
        out[i * 3 + 2] = point[s * 3 + 2];
    }
}

// W row r of the reordered K space -> value (feat rows first, then xyz, then pad)
__device__ __forceinline__ float w_elem(const float* __restrict__ W, int r, int col) {
    if (r < C_IN) return W[(3 + r) * C_OUT + col];
    if (r < 67)   return W[(r - C_IN) * C_OUT + col];
    return 0.0f;
}

// ---------------------------------------------------------------------------
// Phases 1 & 3: gather + f32 WMMA GEMM, double-buffered LDS, B-matrix held in
// registers. STATS=true accumulates per-channel sum/sumsq; STATS=false applies
// BN+ReLU and max-pools over the 16 neighbors.
// Workgroup = 256 threads = 8 waves; wave w owns output channels [16w,16w+16).
// One sample's 16 neighbors == one 16-row A tile (16 x 68), K-steps of 4.
// ---------------------------------------------------------------------------
template <bool STATS>
__global__ __launch_bounds__(256) void td_gemm(const float* __restrict__ point,
                                               const float* __restrict__ feat,
                                               const float* __restrict__ W,
                                               const int* __restrict__ sample_idx,
                                               const int* __restrict__ knn_idx,
                                               float* __restrict__ stats,       // [0:128) sum, [128:256) sumsq
                                               const float* __restrict__ scale, // [128]
                                               const float* __restrict__ shift, // [128]
                                               float* __restrict__ out_feat,    // m x 128
                                               int m) {
    __shared__ float xs[2][KNN * KDIM];  // double buffer, 2 x 4352 B

    const int tid  = threadIdx.x;
    const int wave = tid >> 5;
    const int lane = tid & 31;
    const int half = lane >> 4;          // 0: rows 0-7 / K +0,+1 ; 1: rows 8-15 / K +2,+3
    const int l16  = lane & 15;
    const int koff = half * 2;
    const int col  = (wave << 4) + l16;  // channel owned by this lane in B/C frags

    // Hoist the entire B matrix slice for this lane into registers (34 floats).
    // W is 34 KB -> L2-resident; loaded once per workgroup.
    v2f bfrag[KSTEPS];
#pragma unroll
    for (int ks = 0; ks < KSTEPS; ++ks) {
        const int kb = 4 * ks + koff;
        bfrag[ks].x = w_elem(W, kb, col);
        bfrag[ks].y = w_elem(W, kb + 1, col);
    }

    float lsum = 0.f, lsq = 0.f;
    float sc = 0.f, sh = 0.f;
    if (!STATS) { sc = scale[col]; sh = shift[col]; }

    const int s0    = blockIdx.x * SPW;
    const int nsamp = (m - s0 < SPW) ? (m - s0) : SPW;   // uniform across workgroup

    // Gather-to-register staging (each thread owns quad q of neighbor row j).
    const int j = tid >> 4;              // neighbor row 0..15
    const int q = tid & 15;              // float4 quad 0..15
    float4 gf;                           // 4 feat values -> cols 4q..4q+3
    float  gx = 0.f, gy = 0.f, gz = 0.f; // q==0 thread also carries dxyz

    auto gather = [&](int s) {
        const int nidx = knn_idx[s * KNN + j];
        gf = ((const float4*)(feat + nidx * C_IN))[q];
        if (q == 0) {
            const int sidx = sample_idx[s];
            gx = point[nidx * 3 + 0] - point[sidx * 3 + 0];
            gy = point[nidx * 3 + 1] - point[sidx * 3 + 1];
            gz = point[nidx * 3 + 2] - point[sidx * 3 + 2];
        }
    };
    auto stage = [&](int buf) {
        *(float4*)&xs[buf][j * KDIM + 4 * q] = gf;
        if (q == 0) {
            xs[buf][j * KDIM + 64] = gx;
            xs[buf][j * KDIM + 65] = gy;
            xs[buf][j * KDIM + 66] = gz;
            xs[buf][j * KDIM + 67] = 0.f;
        }
    };

    // Prologue: fill buffer 0 with the first sample.
    gather(s0);
    stage(0);
    __syncthreads();

    for (int si = 0; si < nsamp; ++si) {
        const int s   = s0 + si;
        const int cur = si & 1;
        const bool more = (si + 1 < nsamp);

        // Issue next sample's global loads early; latency hides under WMMAs.
        if (more) gather(s + 1);

        // 16x16 wave tile, K = 68 -> 17 chained f32 WMMAs, B from registers.
        v8f acc = {};
#pragma unroll
        for (int ks = 0; ks < KSTEPS; ++ks) {
            const int kb = 4 * ks + koff;
            v2f a = *(const v2f*)&xs[cur][l16 * KDIM + kb];   // A 16x4 f32 frag
            acc = __builtin_amdgcn_wmma_f32_16x16x4_f32(
                false, a, false, bfrag[ks], (short)0, acc, false, false);
        }

        // Stage next sample into the alternate buffer (no conflict with 'cur').
        if (more) stage(cur ^ 1);

        if (STATS) {
#pragma unroll
            for (int i = 0; i < 8; ++i) { float v = acc[i]; lsum += v; lsq += v * v; }
        } else {
            float mx = 0.f;              // ReLU output >= 0, so 0 is the identity
#pragma unroll
            for (int i = 0; i < 8; ++i)
                mx = fmaxf(mx, fmaxf(sc * acc[i] + sh, 0.f));
            mx = fmaxf(mx, __shfl_xor(mx, 16, 32));           // other 8 neighbor rows
            if (half == 0) out_feat[s * C_OUT + col] = mx;
        }

        __syncthreads();                 // next-buffer stores visible; 'cur' reads done
    }

    if (STATS) {
        lsum += __shfl_xor(lsum, 16, 32);
        lsq  += __shfl_xor(lsq, 16, 32);
        if (half == 0) {                 // each (wave,l16) owns a unique channel
            atomicAdd(&stats[col], lsum);
            atomicAdd(&stats[C_OUT + col], lsq);
        }
    }
}

// ---------------------------------------------------------------------------
// Phase 2: fold stats into per-channel scale/shift.
// ---------------------------------------------------------------------------
__global__ void td_finalize(const float* __restrict__ gamma,
                            const float* __restrict__ beta,
                            float* __restrict__ ws, float invN) {
    int c = threadIdx.x;                 // 128 threads
    float mean = ws[c] * invN;
    float var  = ws[C_OUT + c] * invN - mean * mean;
    float sc   = gamma[c] * rsqrtf(var + BN_EPS);
    ws[256 + c] = sc;
    ws[384 + c] = beta[c] - mean * sc;
}

// ---------------------------------------------------------------------------
extern "C" void kernel_launch(void* const* d_in, const int* in_sizes, int n_in,
                              void* d_out, int out_size, void* d_ws, size_t ws_size,
                              hipStream_t stream) {
    const float* point      = (const float*)d_in[0];
    const float* feat       = (const float*)d_in[1];
    const float* W          = (const float*)d_in[2];
    const float* gamma      = (const float*)d_in[3];
    const float* beta       = (const float*)d_in[4];
    const int*   sample_idx = (const int*)d_in[5];
    const int*   knn_idx    = (const int*)d_in[6];

    const int m = in_sizes[5];           // 60000
    float* out = (float*)d_out;          // [m*3 new_point | m*128 out_feat]
    float* ws  = (float*)d_ws;           // [sum 128 | sumsq 128 | scale 128 | shift 128]

    hipMemsetAsync(ws, 0, 256 * sizeof(float), stream);   // zero stats each launch

    td_newpoint<<<(m + 255) / 256, 256, 0, stream>>>(point, sample_idx, out, m);

    const int nblk = (m + SPW - 1) / SPW;                 // 1875
    td_gemm<true><<<nblk, 256, 0, stream>>>(point, feat, W, sample_idx, knn_idx,
                                            ws, nullptr, nullptr, nullptr, m);

    td_finalize<<<1, C_OUT, 0, stream>>>(gamma, beta, ws, 1.0f / (m * 16.0f));

    td_gemm<false><<<nblk, 256, 0, stream>>>(point, feat, W, sample_idx, knn_idx,
                                             nullptr, ws + 256, ws + 384,
                                             out + (size_t)m * 3, m);
}